// HybridSelfAttention_65481071407402
// MI455X (gfx1250) — compile-verified
//
#include <hip/hip_runtime.h>

typedef float v2f __attribute__((ext_vector_type(2)));
typedef float v8f __attribute__((ext_vector_type(8)));

#define LL 128
#define DD 128
#define XS 132   // padded LDS stride for X (bank-conflict avoidance)
#define SS 132   // padded LDS stride for scores

__global__ __launch_bounds__(256) void hybrid_attn_kernel(
    const float* __restrict__ inputs,   // (B, 128, 128)
    const float* __restrict__ rot_p,    // (384,)
    const float* __restrict__ ent_p,    // (384,)
    const float* __restrict__ conv_w,   // (4,)
    const float* __restrict__ conv_b,   // (1,)
    float* __restrict__ out)            // (B, 128, 128)
{
    __shared__ float Xs[LL * XS];       // inputs[b]
    __shared__ float S [LL * SS];       // softmax scores
    __shared__ float4 qv[LL];           // q0,q1,q2, |q|^2
    __shared__ float4 kv[LL];           // k0,k1,k2, |k|^2
    __shared__ float  sRot[3 * DD];
    __shared__ float  sEnt[3 * DD];
    __shared__ float  red[256];
    __shared__ float  sConv;

    const int tid = threadIdx.x;
    const int b   = blockIdx.x;
    const float* xg = inputs + (size_t)b * (LL * DD);

    // ---- Phase 0: stage X into LDS with async global->LDS b128 copies ----
    // (GLOBAL_LOAD_ASYNC_TO_LDS_B128, GVS mode: lds-addr VGPR, 32-bit byte
    //  offset VGPR, 64-bit SGPR base; tracked by ASYNCcnt)
    #pragma unroll
    for (int it = 0; it < 16; ++it) {
        int f   = tid + it * 256;       // float4 index within the 128x128 tile
        int row = f >> 5;               // (f*4)/128
        int col = (f & 31) << 2;        // (f*4)%128
        unsigned ldsaddr = (unsigned)(uintptr_t)(&Xs[row * XS + col]);
        unsigned goff    = (unsigned)f * 16u;   // byte offset from xg
        asm volatile("global_load_async_to_lds_b128 %0, %1, %2"
                     :
                     : "v"(ldsaddr), "v"(goff), "s"(xg)
                     : "memory");
    }
    for (int i = tid; i < 3 * DD; i += 256) {
        sRot[i] = rot_p[i];
        sEnt[i] = ent_p[i];
    }
    asm volatile("s_wait_asynccnt 0" ::: "memory");
    __syncthreads();

    // ---- Phase 1a: conv partial sums (2x2 VALID conv over 127x127) ----
    const float w00 = conv_w[0], w01 = conv_w[1], w10 = conv_w[2], w11 = conv_w[3];
    float cacc = 0.f;
    for (int p = tid; p < 127 * 127; p += 256) {
        int i = p / 127;
        int j = p - i * 127;
        const float* r0 = &Xs[i * XS + j];
        const float* r1 = r0 + XS;
        cacc += w00 * r0[0] + w01 * r0[1] + w10 * r1[0] + w11 * r1[1];
    }
    red[tid] = cacc;

    // ---- Phase 1b: q/k projections (threads 0..127, one row each) ----
    if (tid < LL) {
        float q0 = 0.f, q1 = 0.f, q2 = 0.f;
        float k0 = 0.f, k1 = 0.f, k2 = 0.f;
        const float* xr = &Xs[tid * XS];
        for (int d = 0; d < DD; ++d) {
            float x = xr[d];
            q0 += x * sRot[3 * d + 0];
            q1 += x * sRot[3 * d + 1];
            q2 += x * sRot[3 * d + 2];
            k0 += x * sEnt[3 * d + 0];
            k1 += x * sEnt[3 * d + 1];
            k2 += x * sEnt[3 * d + 2];
        }
        qv[tid] = make_float4(q0, q1, q2, q0 * q0 + q1 * q1 + q2 * q2);
        kv[tid] = make_float4(k0, k1, k2, k0 * k0 + k1 * k1 + k2 * k2);
    }
    __syncthreads();

    // ---- Phase 1c: tree-reduce conv sum ----
    for (int off = 128; off > 0; off >>= 1) {
        if (tid < off) red[tid] += red[tid + off];
        __syncthreads();
    }
    if (tid == 0) sConv = red[0] / (127.f * 127.f) + conv_b[0];
    // (red[] is reused below; only thread 0 touches red[0] here, and it reads
    //  it before any thread's phase-2 write to red[tid] can alias it.)

    // ---- Phase 2: RBF sim + softmax, all 256 threads (2 threads per row) ----
    {
        const int r    = tid & 127;        // row
        const int hseg = tid >> 7;         // 0: j in [0,64), 1: j in [64,128)
        const int j0   = hseg * 64;
        float4 q = qv[r];
        float* srow = &S[r * SS];
        float psum = 0.f;
        for (int j = j0; j < j0 + 64; ++j) {
            float4 kj   = kv[j];                                   // LDS broadcast
            float dot   = q.x * kj.x + q.y * kj.y + q.z * kj.z;
            float dist2 = q.w + kj.w - 2.f * dot;
            float sim   = __expf(-dist2);                          // GAMMA = 1
            float e     = __expf(sim);                             // sim<=1: no max-sub needed
            srow[j] = e;
            psum   += e;
        }
        red[tid] = psum;
        __syncthreads();
        float rinv = 1.f / (red[r] + red[r + 128]);
        for (int j = j0; j < j0 + 64; ++j) srow[j] *= rinv;
    }
    __syncthreads();

    // ---- Phase 3: attn_out = S @ X via V_WMMA_F32_16X16X4_F32 ----
    // Wave w owns output rows [16w, 16w+16); 8 accumulator tiles across N.
    const int lane = tid & 31;
    const int wave = tid >> 5;
    const int half = lane >> 4;   // 0: lanes 0-15, 1: lanes 16-31
    const int idx  = lane & 15;
    const int mb   = wave * 16;

    v8f acc[8];
    #pragma unroll
    for (int nt = 0; nt < 8; ++nt)
        #pragma unroll
        for (int r = 0; r < 8; ++r) acc[nt][r] = 0.f;

    for (int k0 = 0; k0 < LL; k0 += 4) {
        const int kc = k0 + 2 * half;
        // A fragment: A[M=idx][K=2*half + j], j=0,1 -> one b64 LDS load
        v2f a = *(const v2f*)&S[(mb + idx) * SS + kc];
        #pragma unroll
        for (int nt = 0; nt < 8; ++nt) {
            const int n0 = nt * 16;
            // B fragment: B[K=2*half + j][N=idx]
            v2f bf;
            bf.x = Xs[kc * XS + n0 + idx];
            bf.y = Xs[(kc + 1) * XS + n0 + idx];
            acc[nt] = __builtin_amdgcn_wmma_f32_16x16x4_f32(
                false, a, false, bf, (short)0, acc[nt], false, false);
        }
    }

    const float cm = sConv;
    float* og = out + (size_t)b * (LL * DD);
    #pragma unroll
    for (int nt = 0; nt < 8; ++nt) {
        const int n0 = nt * 16;
        #pragma unroll
        for (int r = 0; r < 8; ++r) {
            // C/D layout: VGPR r holds M = r + 8*half, N = idx
            og[(mb + r + 8 * half) * DD + n0 + idx] = acc[nt][r] + cm;
        }
    }
}

extern "C" void kernel_launch(void* const* d_in, const int* in_sizes, int n_in,
                              void* d_out, int out_size, void* d_ws, size_t ws_size,
                              hipStream_t stream) {
    const float* inputs = (const float*)d_in[0];
    const float* rot    = (const float*)d_in[1];
    const float* ent    = (const float*)d_in[2];
    const float* cw     = (const float*)d_in[3];
    const float* cb     = (const float*)d_in[4];
    float* out          = (float*)d_out;

    const int B = in_sizes[0] / (LL * DD);   // 2048 for the reference shapes
    hybrid_attn_kernel<<<dim3(B), dim3(256), 0, stream>>>(inputs, rot, ent, cw, cb, out);
}